// GCN_670014898796
// MI455X (gfx1250) — compile-verified
//
#include <hip/hip_runtime.h>
#include <math.h>

typedef float v2f __attribute__((ext_vector_type(2)));
typedef float v8f __attribute__((ext_vector_type(8)));

// ---------------------------------------------------------------------------
// Degree / normalization kernels
// ---------------------------------------------------------------------------
__global__ void gcn_init_deg(float* __restrict__ deg, int n) {
    int i = blockIdx.x * blockDim.x + threadIdx.x;
    if (i < n) deg[i] = 1.0f;  // self-loop contributes 1 to deg(A+I)
}

// dst64 points at int64 dst indices; we read only the low 32 bits (values < N)
__global__ void gcn_deg_edges(const int* __restrict__ dst32,
                              float* __restrict__ deg, int E) {
    int e = blockIdx.x * blockDim.x + threadIdx.x;
    if (e < E) {
        int d = dst32[2 * e];
        __hip_atomic_fetch_add(&deg[d], 1.0f, __ATOMIC_RELAXED,
                               __HIP_MEMORY_SCOPE_AGENT);
    }
}

__global__ void gcn_rsqrt(float* __restrict__ deg, int n) {
    int i = blockIdx.x * blockDim.x + threadIdx.x;
    if (i < n) deg[i] = rsqrtf(deg[i]);  // deg >= 1 always
}

// ---------------------------------------------------------------------------
// Dense transform  H[M,NC] = (relu?)A[M,K] @ W[K,NC]  via V_WMMA_F32_16X16X4_F32
// Each wave owns TWO 16x16 output tiles stacked in M (32 rows), so every
// B fragment load feeds two WMMAs. blockDim = (32, NC/16); the 32 x K A-tile
// is staged in LDS once per block and shared by all waves. No bounds checks:
// m0 is clamped so the block always covers valid rows (overlap tiles simply
// recompute identical values).
// ---------------------------------------------------------------------------
template <int K, int NC, bool RELU_A>
__global__ __launch_bounds__(32 * (NC / 16))
void gcn_gemm_wmma(const float* __restrict__ A, const float* __restrict__ W,
                   float* __restrict__ H, int M) {
    constexpr int LDA = K + 4;            // pad stride: avoid 64-bank conflicts
    __shared__ float sA[32 * LDA];

    int m0 = blockIdx.x * 32;
    if (m0 + 32 > M) m0 = M - 32;         // overlap-safe clamp (M >= 32)

    const int lane = threadIdx.x;          // 0..31
    const int wave = threadIdx.y;          // 0..NC/16-1
    const int tid  = wave * 32 + lane;
    constexpr int NT = 32 * (NC / 16);

    // cooperative A-tile load (+ fused ReLU for layer 2), no guards needed
    for (int i = tid; i < 32 * K; i += NT) {
        int r = i / K, k = i - r * K;
        float v = A[(size_t)(m0 + r) * K + k];
        if (RELU_A) v = fmaxf(v, 0.0f);
        sA[r * LDA + k] = v;
    }
    __syncthreads();

    // fp32 WMMA fragment indexing (16x4 A, 4x16 B, 16x16 C):
    //   lanes 0-15 carry K pair {k,k+1}, lanes 16-31 carry {k+2,k+3}
    const int mr = lane & 15;              // A row within tile
    const int kb = (lane >> 4) * 2;        // K sub-offset for this half-wave
    const int nc = wave * 16 + (lane & 15);
    const float* Wp = W + nc;              // column base; k offsets are constant

    v8f acc0 = {0.f, 0.f, 0.f, 0.f, 0.f, 0.f, 0.f, 0.f};
    v8f acc1 = {0.f, 0.f, 0.f, 0.f, 0.f, 0.f, 0.f, 0.f};

#pragma unroll 8
    for (int k = 0; k < K; k += 4) {
        v2f b, a0, a1;
        b.x  = Wp[(size_t)(k + kb) * NC];
        b.y  = Wp[(size_t)(k + kb + 1) * NC];
        a0.x = sA[mr * LDA + k + kb];
        a0.y = sA[mr * LDA + k + kb + 1];
        a1.x = sA[(mr + 16) * LDA + k + kb];
        a1.y = sA[(mr + 16) * LDA + k + kb + 1];
        acc0 = __builtin_amdgcn_wmma_f32_16x16x4_f32(
            false, a0, false, b, (short)0, acc0, false, false);
        acc1 = __builtin_amdgcn_wmma_f32_16x16x4_f32(
            false, a1, false, b, (short)0, acc1, false, false);
    }

    // C/D layout: VGPR v -> M = v (lanes 0-15) or v+8 (lanes 16-31)
    const int rbase = m0 + ((lane >> 4) * 8);
    float* hp0 = H + (size_t)rbase * NC + nc;
    float* hp1 = hp0 + (size_t)16 * NC;
#pragma unroll
    for (int v = 0; v < 8; ++v) {
        hp0[v * NC] = acc0[v];
        hp1[v * NC] = acc1[v];
    }
}

// ---------------------------------------------------------------------------
// out[n,:] = bias + dinv[n]^2 * h[n,:]   (self-loop term; also clears poison)
// ---------------------------------------------------------------------------
template <int NC>
__global__ void gcn_agg_init(const float* __restrict__ h,
                             const float* __restrict__ dinv,
                             const float* __restrict__ bias,
                             float* __restrict__ out, int n) {
    constexpr int G = NC / 4;
    int i = blockIdx.x * blockDim.x + threadIdx.x;
    if (i >= n * G) return;
    int node = i / G;
    int c4   = (i - node * G) * 4;
    float w = dinv[node] * dinv[node];
    const float4 hv = *(const float4*)(h + (unsigned)node * NC + c4);
    const float4 bv = *(const float4*)(bias + c4);
    float4 o;
    o.x = fmaf(w, hv.x, bv.x);
    o.y = fmaf(w, hv.y, bv.y);
    o.z = fmaf(w, hv.z, bv.z);
    o.w = fmaf(w, hv.w, bv.w);
    *(float4*)(out + (unsigned)node * NC + c4) = o;
}

// ---------------------------------------------------------------------------
// out[dst,:] += dinv[src]*dinv[dst] * h[src,:]   (edge scatter, 4-ch vectors)
// ei32 = int64 edge_index reinterpreted as int32 pairs; low word = value.
// ---------------------------------------------------------------------------
template <int NC>
__global__ void gcn_agg_edges(const int* __restrict__ src32,
                              const int* __restrict__ dst32,
                              const float* __restrict__ dinv,
                              const float* __restrict__ h,
                              float* __restrict__ out, int E) {
    constexpr int G = NC / 4;
    unsigned tid = blockIdx.x * blockDim.x + threadIdx.x;  // < E*G < 2^31
    if (tid >= (unsigned)E * G) return;
    unsigned e  = tid / G;
    unsigned c4 = (tid - e * G) * 4;
    int s = src32[2 * e];
    int d = dst32[2 * e];
    float w = dinv[s] * dinv[d];
    const float4 hv = *(const float4*)(h + (unsigned)s * NC + c4);
    float* op = out + (unsigned)d * NC + c4;
    __hip_atomic_fetch_add(op + 0, w * hv.x, __ATOMIC_RELAXED, __HIP_MEMORY_SCOPE_AGENT);
    __hip_atomic_fetch_add(op + 1, w * hv.y, __ATOMIC_RELAXED, __HIP_MEMORY_SCOPE_AGENT);
    __hip_atomic_fetch_add(op + 2, w * hv.z, __ATOMIC_RELAXED, __HIP_MEMORY_SCOPE_AGENT);
    __hip_atomic_fetch_add(op + 3, w * hv.w, __ATOMIC_RELAXED, __HIP_MEMORY_SCOPE_AGENT);
}

// ---------------------------------------------------------------------------
extern "C" void kernel_launch(void* const* d_in, const int* in_sizes, int n_in,
                              void* d_out, int out_size, void* d_ws, size_t ws_size,
                              hipStream_t stream) {
    constexpr int IN = 256, HID = 128, OUT = 64;

    const float* x   = (const float*)d_in[0];
    const int*   ei  = (const int*)d_in[1];   // int64 [2,E] viewed as int32 pairs
    const float* W1  = (const float*)d_in[2];
    const float* b1  = (const float*)d_in[3];
    const float* W2  = (const float*)d_in[4];
    const float* b2  = (const float*)d_in[5];
    float*       out = (float*)d_out;

    const int N = in_sizes[0] / IN;   // 50000
    const int E = in_sizes[1] / 2;    // 800000
    const int* srcp = ei;             // int64 row 0
    const int* dstp = ei + 2 * E;     // int64 row 1 (2 int32 words per elem)

    // workspace carve-up (256B aligned)
    auto align_up = [](size_t v) { return (v + 255) & ~(size_t)255; };
    char* ws = (char*)d_ws;
    float* dinv = (float*)ws;
    size_t off = align_up((size_t)N * sizeof(float));
    float* h1 = (float*)(ws + off);  off += align_up((size_t)N * HID * sizeof(float));
    float* o1 = (float*)(ws + off);  off += align_up((size_t)N * HID * sizeof(float));
    float* h2 = (float*)(ws + off);
    (void)ws_size; (void)n_in; (void)out_size;

    const int T = 256;
    // 1) deg(A+I) over dst, then dinv = rsqrt(deg)
    gcn_init_deg<<<(N + T - 1) / T, T, 0, stream>>>(dinv, N);
    gcn_deg_edges<<<(E + T - 1) / T, T, 0, stream>>>(dstp, dinv, E);
    gcn_rsqrt<<<(N + T - 1) / T, T, 0, stream>>>(dinv, N);

    // 2) layer 1: h1 = x @ W1  (fp32 WMMA, 32 rows per block)
    gcn_gemm_wmma<IN, HID, false>
        <<<(N + 31) / 32, dim3(32, HID / 16), 0, stream>>>(x, W1, h1, N);

    // 3) layer 1 aggregate: o1 = b1 + D^-1/2 (A+I) D^-1/2 h1
    {
        int wi = N * (HID / 4);
        gcn_agg_init<HID><<<(wi + T - 1) / T, T, 0, stream>>>(h1, dinv, b1, o1, N);
        int we = E * (HID / 4) / T + 1;
        gcn_agg_edges<HID><<<we, T, 0, stream>>>(srcp, dstp, dinv, h1, o1, E);
    }

    // 4) layer 2: h2 = relu(o1) @ W2  (ReLU fused into A-tile load)
    gcn_gemm_wmma<HID, OUT, true>
        <<<(N + 31) / 32, dim3(32, OUT / 16), 0, stream>>>(o1, W2, h2, N);

    // 5) layer 2 aggregate into d_out
    {
        int wi = N * (OUT / 4);
        gcn_agg_init<OUT><<<(wi + T - 1) / T, T, 0, stream>>>(h2, dinv, b2, out, N);
        int we = E * (OUT / 4) / T + 1;
        gcn_agg_edges<OUT><<<we, T, 0, stream>>>(srcp, dstp, dinv, h2, out, E);
    }
}